// NameGCN_4956392259829
// MI455X (gfx1250) — compile-verified
//
#include <hip/hip_runtime.h>

#define DIM 128
#define NENT 100000
#define NEDGE 500000
#define NSEED 10000

typedef __attribute__((ext_vector_type(2))) float v2f;
typedef __attribute__((ext_vector_type(8))) float v8f;

// ---------------- degree ----------------
__global__ void init_deg_kernel(float* __restrict__ deg, int n) {
    int i = blockIdx.x * blockDim.x + threadIdx.x;
    if (i < n) deg[i] = 1.0f;   // self loop
}

__global__ void count_deg_kernel(const int* __restrict__ edges,
                                 float* __restrict__ deg, int nedge) {
    int e = blockIdx.x * blockDim.x + threadIdx.x;
    if (e < nedge) {
        atomicAdd(&deg[edges[2 * e]],     1.0f);
        atomicAdd(&deg[edges[2 * e + 1]], 1.0f);
    }
}

// -------- agg init with self-loop contribution: agg[v] = x[v]/deg[v] --------
__global__ void init_agg_kernel(const float* __restrict__ x,
                                const float* __restrict__ deg,
                                float* __restrict__ agg, int n) {
    int t = blockIdx.x * blockDim.x + threadIdx.x;   // one thread per 4 floats
    int v = t >> 5, lane = t & 31;
    if (v >= n) return;
    float inv = 1.0f / deg[v];                        // rsqrt(deg*deg) == 1/deg
    float4 xv = ((const float4*)(x + (size_t)v * DIM))[lane];
    float4 r  = make_float4(xv.x * inv, xv.y * inv, xv.z * inv, xv.w * inv);
    ((float4*)(agg + (size_t)v * DIM))[lane] = r;
}

// ---------------- SpMM scatter: one wave32 per undirected edge ----------------
__global__ __launch_bounds__(256) void spmm_edges_kernel(
    const int* __restrict__ edges, const float* __restrict__ deg,
    const float* __restrict__ x, float* __restrict__ agg, int nedge) {
    int w    = (blockIdx.x * blockDim.x + threadIdx.x) >> 5;
    int lane = threadIdx.x & 31;
    if (w >= nedge) return;
    int e0 = edges[2 * w];
    int e1 = edges[2 * w + 1];
    float nrm = rsqrtf(deg[e0] * deg[e1]);
    float4 a = ((const float4*)(x + (size_t)e0 * DIM))[lane];
    float4 b = ((const float4*)(x + (size_t)e1 * DIM))[lane];
    float* p1 = agg + (size_t)e1 * DIM + lane * 4;
    float* p0 = agg + (size_t)e0 * DIM + lane * 4;
    atomicAdd(p1 + 0, a.x * nrm);
    atomicAdd(p1 + 1, a.y * nrm);
    atomicAdd(p1 + 2, a.z * nrm);
    atomicAdd(p1 + 3, a.w * nrm);
    atomicAdd(p0 + 0, b.x * nrm);
    atomicAdd(p0 + 1, b.y * nrm);
    atomicAdd(p0 + 2, b.z * nrm);
    atomicAdd(p0 + 3, b.w * nrm);
}

// ------- H = relu(agg @ W + xres), fp32 WMMA 16x16x4, wave per 16x16 tile -------
__global__ __launch_bounds__(256) void gemm_relu_kernel(
    const float* __restrict__ agg, const float* __restrict__ W,
    const float* __restrict__ xres, float* __restrict__ out, int n) {
    __shared__ float sA[16 * 132];          // 16-row A strip, padded stride
    const int row0 = blockIdx.x * 16;
    const int tid  = threadIdx.x;

    // cooperative load of the shared A strip (16 x 128)
    for (int i = tid; i < 16 * 128; i += 256) {
        int r = i >> 7, c = i & 127;
        int gr = row0 + r;
        sA[r * 132 + c] = (gr < n) ? agg[(size_t)gr * DIM + c] : 0.0f;
    }
    __syncthreads();

    const int wave = tid >> 5;              // 0..7 -> 16-col tile
    const int lane = tid & 31;
    const int col0 = wave * 16;
    const int m    = lane & 15;
    const int half = lane >> 4;             // 0: K={0,1}/M low, 1: K={2,3}/M high
    const int koff = half * 2;
    const int ncol = lane & 15;

    v8f c = {};
#pragma unroll
    for (int k0 = 0; k0 < DIM; k0 += 4) {
        v2f a, b;
        a.x = sA[m * 132 + k0 + koff];
        a.y = sA[m * 132 + k0 + koff + 1];
        b.x = W[(size_t)(k0 + koff)     * DIM + col0 + ncol];
        b.y = W[(size_t)(k0 + koff + 1) * DIM + col0 + ncol];
        c = __builtin_amdgcn_wmma_f32_16x16x4_f32(
                /*neg_a=*/false, a, /*neg_b=*/false, b,
                /*c_mod=*/(short)0, c, /*reuse_a=*/false, /*reuse_b=*/false);
    }

#pragma unroll
    for (int r = 0; r < 8; ++r) {
        int gm = row0 + r + half * 8;
        if (gm < n) {
            size_t idx = (size_t)gm * DIM + col0 + ncol;
            float v = c[r] + xres[idx];      // residual; safe in-place (per-element RMW)
            out[idx] = v > 0.0f ? v : 0.0f;
        }
    }
}

// ---------------- seed gather ----------------
__global__ void gather_kernel(const int* __restrict__ seeds,
                              const float* __restrict__ hid,
                              float* __restrict__ out, int nseed) {
    int t = blockIdx.x * blockDim.x + threadIdx.x;
    int s = t >> 5, lane = t & 31;
    if (s >= nseed) return;
    int v = seeds[s];
    ((float4*)(out + (size_t)s * DIM))[lane] =
        ((const float4*)(hid + (size_t)v * DIM))[lane];
}

static void run_graph(const int* seeds, const float* emb, const int* edges,
                      const float* W1, const float* W2,
                      float* deg, float* agg,
                      float* out_ent, float* out_seed, hipStream_t stream) {
    const int B = 256;
    init_deg_kernel<<<(NENT + B - 1) / B, B, 0, stream>>>(deg, NENT);
    count_deg_kernel<<<(NEDGE + B - 1) / B, B, 0, stream>>>(edges, deg, NEDGE);

    // ---- layer 1: h1 = relu(A_norm @ emb @ W1 + emb) -> out_ent ----
    init_agg_kernel<<<(NENT * 32 + B - 1) / B, B, 0, stream>>>(emb, deg, agg, NENT);
    spmm_edges_kernel<<<(NEDGE * 32 + B - 1) / B, B, 0, stream>>>(edges, deg, emb, agg, NEDGE);
    gemm_relu_kernel<<<(NENT + 15) / 16, B, 0, stream>>>(agg, W1, emb, out_ent, NENT);

    // ---- layer 2: h2 = relu(A_norm @ h1 @ W2 + h1) in-place on out_ent ----
    init_agg_kernel<<<(NENT * 32 + B - 1) / B, B, 0, stream>>>(out_ent, deg, agg, NENT);
    spmm_edges_kernel<<<(NEDGE * 32 + B - 1) / B, B, 0, stream>>>(edges, deg, out_ent, agg, NEDGE);
    gemm_relu_kernel<<<(NENT + 15) / 16, B, 0, stream>>>(agg, W2, out_ent, out_ent, NENT);

    gather_kernel<<<(NSEED * 32 + B - 1) / B, B, 0, stream>>>(seeds, out_ent, out_seed, NSEED);
}

extern "C" void kernel_launch(void* const* d_in, const int* in_sizes, int n_in,
                              void* d_out, int out_size, void* d_ws, size_t ws_size,
                              hipStream_t stream) {
    const int*   sr_seeds = (const int*)d_in[0];
    const int*   tg_seeds = (const int*)d_in[1];
    // d_in[2], d_in[3] (triples) unused by the reference
    const float* emb_sr   = (const float*)d_in[4];
    const float* emb_tg   = (const float*)d_in[5];
    const int*   edges_sr = (const int*)d_in[6];
    const int*   edges_tg = (const int*)d_in[7];
    const float* W1       = (const float*)d_in[8];
    const float* W2       = (const float*)d_in[9];

    float* out = (float*)d_out;
    float* out_seed_sr = out;                               // [NSEED,128]
    float* out_seed_tg = out + (size_t)NSEED * DIM;         // [NSEED,128]
    float* out_ent_sr  = out + (size_t)2 * NSEED * DIM;     // [NENT,128]
    float* out_ent_tg  = out_ent_sr + (size_t)NENT * DIM;   // [NENT,128]

    // workspace: deg [NENT] + agg [NENT*DIM], reused sequentially per graph
    float* deg = (float*)d_ws;
    float* agg = deg + ((NENT + 255) & ~255);

    run_graph(sr_seeds, emb_sr, edges_sr, W1, W2, deg, agg,
              out_ent_sr, out_seed_sr, stream);
    run_graph(tg_seeds, emb_tg, edges_tg, W1, W2, deg, agg,
              out_ent_tg, out_seed_tg, stream);
}